// ChunkLayer_773094113544
// MI455X (gfx1250) — compile-verified
//
#include <hip/hip_runtime.h>
#include <hip/hip_bf16.h>

#define S_LEN 8192
#define DIM   1024
#define BATCH 4

// ---------------------------------------------------------------------------
// Kernel 1: per-batch stable partition scan.
// One 1024-thread block per batch. Computes, for every position, its rank
// among boundary positions (exclusive prefix count) via wave32 ballot/popc +
// cross-wave LDS scan, then scatters the selection indices:
//   boundary pos  -> slot = excl_rank
//   other pos     -> slot = total_boundaries + (pos - excl_rank)
// which reproduces jnp.argsort(token_idx, stable=True)[:, :max_chunks].
// Also writes num_tokens (as float) into the tail of d_out.
// ---------------------------------------------------------------------------
__global__ __launch_bounds__(1024)
void chunk_scan_kernel(const int* __restrict__ boundaries,
                       int* __restrict__ sel,
                       float* __restrict__ num_tokens_out,
                       int max_chunks)
{
    const int b = blockIdx.x;
    const int* bnd = boundaries + (size_t)b * S_LEN;
    int* selb = sel + (size_t)b * max_chunks;

    __shared__ int wave_sums[32];
    __shared__ int wave_off[33];
    __shared__ int sh_total;
    __shared__ int sh_tilebase;

    const int tid  = threadIdx.x;
    const int lane = tid & 31;          // wave32 on gfx1250
    const int wv   = tid >> 5;
    const int nw   = blockDim.x >> 5;   // 32 waves

    // ---- pass 1: total boundary count for this batch ----
    int cnt = 0;
    for (int i = tid; i < S_LEN; i += blockDim.x) cnt += (bnd[i] != 0);
    for (int off = 16; off > 0; off >>= 1) cnt += __shfl_down(cnt, off, 32);
    if (lane == 0) wave_sums[wv] = cnt;
    __syncthreads();
    if (tid == 0) {
        int t = 0;
        for (int w = 0; w < nw; ++w) t += wave_sums[w];
        sh_total = t;
        sh_tilebase = 0;
        num_tokens_out[b] = (float)t;   // int count, exactly representable
    }
    __syncthreads();
    const int total = sh_total;

    // ---- pass 2: tiled exclusive scan + scatter of selection indices ----
    for (int t0 = 0; t0 < S_LEN; t0 += blockDim.x) {
        const int pos  = t0 + tid;                       // S_LEN % blockDim == 0
        const int pred = (bnd[pos] != 0);
        const unsigned mask = (unsigned)__ballot(pred);  // wave32 mask in low bits
        const int rank_in_wave = __popc(mask & ((1u << lane) - 1u));
        const int wave_total   = __popc(mask);
        if (lane == 0) wave_sums[wv] = wave_total;
        __syncthreads();
        if (tid == 0) {
            int acc = 0;
            for (int w = 0; w < nw; ++w) { wave_off[w] = acc; acc += wave_sums[w]; }
            wave_off[nw] = acc;                          // tile total
        }
        __syncthreads();
        const int excl = sh_tilebase + wave_off[wv] + rank_in_wave;
        const int slot = pred ? excl : (total + (pos - excl));
        if (slot < max_chunks) selb[slot] = pos;
        __syncthreads();
        if (tid == 0) sh_tilebase += wave_off[nw];
        __syncthreads();
    }
}

// ---------------------------------------------------------------------------
// Kernel 2: HBM-saturating row gather via the CDNA5 async Global<->LDS DMA
// path. One 256-thread block moves one 4KB row: each lane DMAs a disjoint
// 16B slice memory->LDS (no VGPR staging), waits ASYNCcnt, then DMAs the same
// slice LDS->memory. Lanes own disjoint LDS slices, so only the per-wave
// s_wait_asynccnt ordering is required (no barriers).
// ---------------------------------------------------------------------------
__global__ __launch_bounds__(256)
void chunk_gather_kernel(const float* __restrict__ x,
                         const int* __restrict__ sel,
                         float* __restrict__ out,
                         int max_chunks)
{
    const int j = blockIdx.x;                 // chunk slot
    const int b = blockIdx.y;                 // batch
    const int row = sel[(size_t)b * max_chunks + j];

    const float* src = x   + ((size_t)b * S_LEN      + row) * DIM;
    float*       dst = out + ((size_t)b * max_chunks + j  ) * DIM;

    __shared__ float buf[DIM];                // 4 KB staging tile in LDS

    const int tid = threadIdx.x;
    // Low 32 bits of a generic pointer to LDS == wave-relative LDS byte
    // address (ISA 10.2: LDS aperture -> addr[31:0]); that is what the async
    // instructions' LDS-address VGPR expects.
    const unsigned lds_addr =
        (unsigned)(size_t)(void*)&buf[0] + (unsigned)(tid * 16);
    const float* gsrc = src + tid * 4;        // 16 B per lane
    float*       gdst = dst + tid * 4;

    // memory -> LDS (ASYNCcnt++), tracked per-wave
    asm volatile("global_load_async_to_lds_b128 %0, %1, off"
                 :: "v"(lds_addr), "v"(gsrc) : "memory");
    asm volatile("s_wait_asynccnt 0" ::: "memory");
    // LDS -> memory
    asm volatile("global_store_async_from_lds_b128 %0, %1, off"
                 :: "v"(gdst), "v"(lds_addr) : "memory");
    asm volatile("s_wait_asynccnt 0" ::: "memory");
}

// ---------------------------------------------------------------------------
extern "C" void kernel_launch(void* const* d_in, const int* in_sizes, int n_in,
                              void* d_out, int out_size, void* d_ws, size_t ws_size,
                              hipStream_t stream)
{
    const float* x          = (const float*)d_in[0];
    const int*   boundaries = (const int*)d_in[1];
    float*       out        = (float*)d_out;

    const int B = BATCH;
    // out_size = B*max_chunks*DIM + B  (gathered rows + per-batch counts)
    const int max_chunks = (out_size - B) / (B * DIM);

    int*   sel            = (int*)d_ws;                         // B*max_chunks ints
    float* num_tokens_out = out + (size_t)B * max_chunks * DIM; // tail of d_out

    chunk_scan_kernel<<<dim3(B), dim3(1024), 0, stream>>>(
        boundaries, sel, num_tokens_out, max_chunks);

    chunk_gather_kernel<<<dim3(max_chunks, B), dim3(256), 0, stream>>>(
        x, sel, out, max_chunks);
}